// Attention_9483287790338
// MI455X (gfx1250) — compile-verified
//
#include <hip/hip_runtime.h>
#include <hip/hip_bf16.h>

// ---------------- problem constants (exact divisors, no bounds checks) -----
#define BATCH   16
#define CIN     256
#define NPIX    4096      // 64*64
#define OC3     768       // 3*HEADS*DHEAD
#define NHEADS  4
#define DHEAD   64
#define QSCALE  0.125f    // DHEAD^-0.5
#define LN_EPS  1e-5f

typedef float v2f __attribute__((ext_vector_type(2)));
typedef float v8f __attribute__((ext_vector_type(8)));

// D(16x16,f32) = A(16x4,f32) * B(4x16,f32) + C   --- V_WMMA_F32_16X16X4_F32
static __device__ __forceinline__ v8f wmma4(v2f a, v2f b, v8f c) {
    return __builtin_amdgcn_wmma_f32_16x16x4_f32(
        /*neg_a=*/false, a, /*neg_b=*/false, b,
        /*c_mod=*/(short)0, c, /*reuse_a=*/false, /*reuse_b=*/false);
}

// ===========================================================================
// K1: qkv[b][o][n] = sum_c Wqkv[o][c] * X[b][c][n]
// grid (NPIX/32, OC3/64, BATCH), block 256 (8 waves: 4M x 2N of 16x16 tiles)
// ===========================================================================
__global__ __launch_bounds__(256) void qkv_gemm_k(
    const float* __restrict__ X, const float* __restrict__ W,
    float* __restrict__ QKV)
{
    const int t = threadIdx.x, lane = t & 31, w = t >> 5;
    const int wm = w >> 1, wn = w & 1;
    const int b  = blockIdx.z;
    const int Mb = blockIdx.y * 64;
    const int Nb = blockIdx.x * 32;

    __shared__ float Wt[64 * 17];   // padded, coprime with 64 banks
    __shared__ float Xt[16 * 33];

    const float* Xb = X + (size_t)b * CIN * NPIX;
    const int lo  = lane & 15;
    const int ko  = (lane >= 16) ? 2 : 0;
    const int hi8 = (lane >= 16) ? 8 : 0;
    v8f acc = {};

    for (int k0 = 0; k0 < CIN; k0 += 16) {
        // stage W tile 64x16 (one float4 per thread, coalesced)
        {
            const int m  = t >> 2;
            const int kk = (t & 3) * 4;
            const float4 wv =
                *(const float4*)(W + (size_t)(Mb + m) * CIN + k0 + kk);
            Wt[m*17 + kk+0] = wv.x; Wt[m*17 + kk+1] = wv.y;
            Wt[m*17 + kk+2] = wv.z; Wt[m*17 + kk+3] = wv.w;
        }
        // stage X tile 16x32 (coalesced 32-wide)
        for (int i = 0; i < 2; ++i) {
            const int idx = t + i*256;
            const int kk = idx >> 5, nn = idx & 31;
            Xt[kk*33 + nn] = Xb[(size_t)(k0 + kk) * NPIX + Nb + nn];
        }
        if (k0 + 16 < CIN)  // gfx1250 global_prefetch_b8 of next X chunk
            __builtin_prefetch(Xb + (size_t)(k0 + 16) * NPIX + Nb + lane, 0, 1);
        __syncthreads();
        for (int kk = 0; kk < 16; kk += 4) {
            v2f a, bb;
            a.x  = Wt[(wm*16 + lo)*17 + kk + ko];
            a.y  = Wt[(wm*16 + lo)*17 + kk + ko + 1];
            bb.x = Xt[(kk + ko    )*33 + wn*16 + lo];
            bb.y = Xt[(kk + ko + 1)*33 + wn*16 + lo];
            acc = wmma4(a, bb, acc);
        }
        __syncthreads();
    }
    float* out = QKV + (size_t)b * OC3 * NPIX;
    for (int r = 0; r < 8; ++r) {
        const int row = Mb + wm*16 + r + hi8;
        const int col = Nb + wn*16 + lo;
        out[(size_t)row * NPIX + col] = acc[r];
    }
}

// ===========================================================================
// K2: per-row (b,h,d) stats of k over n: max and sum(exp(x-max))
// grid BATCH*NHEADS*DHEAD, block 256
// ===========================================================================
__global__ __launch_bounds__(256) void kstats_k(
    const float* __restrict__ QKV, float* __restrict__ kmax,
    float* __restrict__ ksum)
{
    __shared__ float red[256];
    const int t = threadIdx.x;
    const int r = blockIdx.x;                 // b*256 + h*64 + d
    const int b = r >> 8;
    const float* row = QKV + ((size_t)b * OC3 + 256 + (r & 255)) * NPIX;

    float m = -3.4e38f;
    for (int i = t; i < NPIX; i += 256) m = fmaxf(m, row[i]);
    red[t] = m; __syncthreads();
    for (int s = 128; s > 0; s >>= 1) {
        if (t < s) red[t] = fmaxf(red[t], red[t + s]);
        __syncthreads();
    }
    m = red[0]; __syncthreads();

    float s = 0.f;
    for (int i = t; i < NPIX; i += 256) s += __expf(row[i] - m);
    red[t] = s; __syncthreads();
    for (int ss = 128; ss > 0; ss >>= 1) {
        if (t < ss) red[t] += red[t + ss];
        __syncthreads();
    }
    if (t == 0) { kmax[r] = m; ksum[r] = red[0]; }
}

// ===========================================================================
// K3: context[d][e] = sum_n softmax_n(k)[d][n] * (v[e][n]/NPIX)  per (b,h)
// grid BATCH*NHEADS, block 512 (16 waves = the 4x4 tile grid of 64x64)
// ===========================================================================
__global__ __launch_bounds__(512) void context_k(
    const float* __restrict__ QKV, const float* __restrict__ kmax,
    const float* __restrict__ ksum, float* __restrict__ CTX)
{
    const int t = threadIdx.x, lane = t & 31, w = t >> 5;
    const int bh = blockIdx.x, b = bh >> 2, h = bh & 3;

    __shared__ float kt[64 * 33];
    __shared__ float vt[64 * 33];
    __shared__ float smax[64], sinv[64];
    if (t < 64) {
        smax[t] = kmax[bh*64 + t];
        sinv[t] = 1.0f / ksum[bh*64 + t];
    }
    __syncthreads();

    const float* kp = QKV + ((size_t)b * OC3 + 256 + h*64) * NPIX;
    const float* vp = QKV + ((size_t)b * OC3 + 512 + h*64) * NPIX;
    const int d0 = (w >> 2) * 16, e0 = (w & 3) * 16;
    const int lo = lane & 15, ko = (lane >= 16) ? 2 : 0, hi8 = (lane >= 16) ? 8 : 0;
    v8f acc = {};

    for (int n0 = 0; n0 < NPIX; n0 += 32) {
        for (int i = 0; i < 4; ++i) {
            const int idx = t + i*512;
            const int dd = idx >> 5, nn = idx & 31;
            kt[dd*33 + nn] = __expf(kp[(size_t)dd * NPIX + n0 + nn] - smax[dd]) * sinv[dd];
            vt[dd*33 + nn] = vp[(size_t)dd * NPIX + n0 + nn] * (1.0f / (float)NPIX);
        }
        __syncthreads();
        for (int kk = 0; kk < 32; kk += 4) {
            v2f a, bb;
            a.x  = kt[(d0 + lo)*33 + kk + ko];
            a.y  = kt[(d0 + lo)*33 + kk + ko + 1];
            bb.x = vt[(e0 + lo)*33 + kk + ko];
            bb.y = vt[(e0 + lo)*33 + kk + ko + 1];
            acc = wmma4(a, bb, acc);
        }
        __syncthreads();
    }
    for (int r = 0; r < 8; ++r)
        CTX[((size_t)bh*64 + d0 + r + hi8)*64 + e0 + lo] = acc[r];
}

// ===========================================================================
// K4: q-softmax over d (fused, never hits HBM) + out[e][n] = ct^T @ q'
// grid (NPIX/64, NHEADS, BATCH), block 256 (8 waves, 2 tiles each)
// ===========================================================================
__global__ __launch_bounds__(256) void attnout_k(
    const float* __restrict__ QKV, const float* __restrict__ CTX,
    float* __restrict__ OUT)
{
    const int t = threadIdx.x, lane = t & 31, w = t >> 5;
    const int b = blockIdx.z, h = blockIdx.y;
    const int nblk = blockIdx.x * 64;

    __shared__ float ct[64 * 65];     // context, d-major
    __shared__ float qb[64 * 65];     // q block,  d-major

    const float* qp = QKV + ((size_t)b * OC3 + h*64) * NPIX;
    const float* cp = CTX + (size_t)(b*4 + h) * 64 * 64;
    for (int i = 0; i < 16; ++i) {
        const int idx = t + i*256;
        const int dd = idx >> 6, ee = idx & 63;
        ct[dd*65 + ee] = cp[idx];
        qb[dd*65 + ee] = qp[(size_t)dd * NPIX + nblk + ee];
    }
    __syncthreads();

    if (t < 64) {                      // softmax over d, one column per lane
        const int col = t;
        float m = -3.4e38f;
        for (int d = 0; d < 64; ++d) m = fmaxf(m, qb[d*65 + col]);
        float s = 0.f;
        for (int d = 0; d < 64; ++d) s += __expf(qb[d*65 + col] - m);
        const float inv = QSCALE / s;
        for (int d = 0; d < 64; ++d)
            qb[d*65 + col] = __expf(qb[d*65 + col] - m) * inv;
    }
    __syncthreads();

    const int lo = lane & 15, ko = (lane >= 16) ? 2 : 0, hi8 = (lane >= 16) ? 8 : 0;
    for (int tt = w; tt < 16; tt += 8) {     // uniform per wave (2 tiles)
        const int e0 = (tt >> 2) * 16, n0 = (tt & 3) * 16;
        v8f acc = {};
        for (int kk = 0; kk < 64; kk += 4) {
            v2f a, bb;                        // A[M=e][K=d] = ct[d][e]
            a.x  = ct[(kk + ko    )*65 + e0 + lo];
            a.y  = ct[(kk + ko + 1)*65 + e0 + lo];
            bb.x = qb[(kk + ko    )*65 + n0 + lo];
            bb.y = qb[(kk + ko + 1)*65 + n0 + lo];
            acc = wmma4(a, bb, acc);
        }
        for (int r = 0; r < 8; ++r)
            OUT[((size_t)b*256 + h*64 + e0 + r + hi8)*NPIX + nblk + n0 + lo] = acc[r];
    }
}

// ===========================================================================
// K5: y = Wout @ out + bias, then LayerNorm over 256 channels per pixel, * g
// grid (NPIX/64, BATCH), block 256 (8 waves; wave w owns rows [32w,32w+32))
// ===========================================================================
__global__ __launch_bounds__(256) void proj_ln_k(
    const float* __restrict__ OUT, const float* __restrict__ Wout,
    const float* __restrict__ bias, const float* __restrict__ g,
    float* __restrict__ Y)
{
    const int t = threadIdx.x, lane = t & 31, w = t >> 5;
    const int b = blockIdx.y;
    const int nblk = blockIdx.x * 64;

    __shared__ float ot[64 * 65];     // K-chunk of attn-out
    __shared__ float yt[256 * 64];    // full y tile for LN
    __shared__ float bg[512];         // bias | g
    bg[t] = bias[t]; bg[256 + t] = g[t];

    const int lo = lane & 15, ko = (lane >= 16) ? 2 : 0, hi8 = (lane >= 16) ? 8 : 0;
    const int M0 = w * 32;
    v8f acc[2][4] = {};

    for (int c0 = 0; c0 < 256; c0 += 64) {
        __syncthreads();
        for (int i = 0; i < 16; ++i) {
            const int idx = t + i*256;
            const int cc = idx >> 6, nn = idx & 63;
            ot[cc*65 + nn] = OUT[((size_t)b*256 + c0 + cc)*NPIX + nblk + nn];
        }
        __syncthreads();
        for (int kk = 0; kk < 64; kk += 4) {
            v2f a0, a1;
            a0.x = Wout[(size_t)(M0      + lo)*256 + c0 + kk + ko];
            a0.y = Wout[(size_t)(M0      + lo)*256 + c0 + kk + ko + 1];
            a1.x = Wout[(size_t)(M0 + 16 + lo)*256 + c0 + kk + ko];
            a1.y = Wout[(size_t)(M0 + 16 + lo)*256 + c0 + kk + ko + 1];
            for (int nt = 0; nt < 4; ++nt) {
                v2f bb;
                bb.x = ot[(kk + ko    )*65 + nt*16 + lo];
                bb.y = ot[(kk + ko + 1)*65 + nt*16 + lo];
                acc[0][nt] = wmma4(a0, bb, acc[0][nt]);
                acc[1][nt] = wmma4(a1, bb, acc[1][nt]);
            }
        }
    }
    __syncthreads();
    for (int mt = 0; mt < 2; ++mt)
        for (int nt = 0; nt < 4; ++nt)
            for (int r = 0; r < 8; ++r) {
                const int row = M0 + mt*16 + r + hi8;
                yt[row*64 + nt*16 + lo] = acc[mt][nt][r] + bg[row];
            }
    __syncthreads();

    if (t < 64) {                      // LayerNorm: one pixel column per lane
        const int col = t;
        float s = 0.f, s2 = 0.f;
        for (int rr = 0; rr < 256; ++rr) {
            const float v = yt[rr*64 + col];
            s += v; s2 += v * v;
        }
        const float mean = s * (1.0f / 256.0f);
        const float var  = s2 * (1.0f / 256.0f) - mean * mean;
        const float rstd = rsqrtf(var + LN_EPS);
        for (int rr = 0; rr < 256; ++rr) {
            const float v = (yt[rr*64 + col] - mean) * rstd * bg[256 + rr];
            Y[((size_t)b*256 + rr)*NPIX + nblk + col] = v;
        }
    }
}

// ===========================================================================
extern "C" void kernel_launch(void* const* d_in, const int* in_sizes, int n_in,
                              void* d_out, int out_size, void* d_ws, size_t ws_size,
                              hipStream_t stream)
{
    (void)in_sizes; (void)n_in; (void)out_size; (void)ws_size;
    const float* x     = (const float*)d_in[0];
    const float* w_qkv = (const float*)d_in[1];
    const float* w_out = (const float*)d_in[2];
    const float* b_out = (const float*)d_in[3];
    const float* g     = (const float*)d_in[4];
    float* y = (float*)d_out;

    // workspace layout (floats): qkv | kmax | ksum | context | attn_out
    float* qkv   = (float*)d_ws;                                   // 16*768*4096
    float* kmaxp = qkv   + (size_t)BATCH * OC3 * NPIX;             // 4096
    float* ksump = kmaxp + (size_t)BATCH * NHEADS * DHEAD;         // 4096
    float* ctx   = ksump + (size_t)BATCH * NHEADS * DHEAD;         // 16*4*64*64
    float* attn  = ctx   + (size_t)BATCH * NHEADS * DHEAD * DHEAD; // 16*256*4096

    qkv_gemm_k<<<dim3(NPIX/32, OC3/64, BATCH), 256, 0, stream>>>(x, w_qkv, qkv);
    kstats_k  <<<BATCH * NHEADS * DHEAD, 256, 0, stream>>>(qkv, kmaxp, ksump);
    context_k <<<BATCH * NHEADS, 512, 0, stream>>>(qkv, kmaxp, ksump, ctx);
    attnout_k <<<dim3(NPIX/64, NHEADS, BATCH), 256, 0, stream>>>(qkv, ctx, attn);
    proj_ln_k <<<dim3(NPIX/64, BATCH), 256, 0, stream>>>(attn, w_out, b_out, g, y);
}